// GHMCLoss_17987323036120
// MI455X (gfx1250) — compile-verified
//
#include <hip/hip_runtime.h>
#include <stdint.h>

#define BINS   30
#define NCLS   40
#define NBATCH 262144
#define NTOT   (NBATCH * NCLS)   /* 10,485,760 elements */
#define NV     (NTOT / 4)        /* float4 groups; 4 | 40 so a group never crosses a row */
#define MMT    0.6f

// ---------------- workspace layout ----------------
// [0,      4800)  : counts, 1200 x u32
// [4800,   9600)  : wtab,   1200 x f32  (B / (new_acc * n) per (class,bin))
// [16384, +42MB)  : cached bce (f32 per element)        } only if ws_size permits
// [...,   +10.5MB): cached bin (u8 per element, packed) }
static const size_t CACHE_OFF = 16384;

// ---------------- CDNA5 async copy helpers ----------------
__device__ __forceinline__ void async_lds_b128(uint32_t lds_off, const void* gaddr) {
    // GLOBAL_LOAD_ASYNC_TO_LDS_B128: vdst = LDS byte address, vaddr = 64-bit global addr (GV mode)
    asm volatile("global_load_async_to_lds_b128 %0, %1, off"
                 :: "v"(lds_off), "v"(gaddr) : "memory");
}
__device__ __forceinline__ void wait_async0() {
    asm volatile("s_wait_asynccnt 0" ::: "memory");
}

// ---------------- block reduction: wave32 shuffle + tiny cross-wave ----------------
__device__ __forceinline__ void block_reduce_add(float v, float* out, float scale) {
    __shared__ float wsum[8];                 // 256 threads = 8 wave32
#pragma unroll
    for (int off = 16; off > 0; off >>= 1) v += __shfl_down(v, off, 32);
    const int wid  = threadIdx.x >> 5;
    const int lane = threadIdx.x & 31;
    if (lane == 0) wsum[wid] = v;
    __syncthreads();
    if (threadIdx.x < 8) {
        float t = wsum[threadIdx.x];
#pragma unroll
        for (int off = 4; off > 0; off >>= 1) t += __shfl_down(t, off, 8);
        if (threadIdx.x == 0) atomicAdd(out, t * scale);   // global_atomic_add_f32
    }
}

// ---------------- kernel A: zero counts + output ----------------
__global__ void k_zero(uint32_t* __restrict__ counts, float* __restrict__ out) {
    int g = blockIdx.x * blockDim.x + threadIdx.x;
    if (g < NCLS * BINS) counts[g] = 0u;
    if (g == 0) out[0] = 0.0f;
}

// ---------------- kernel B: histogram (+ optional bce/bin cache) ----------------
template <bool CACHE>
__global__ void k_hist(const float4* __restrict__ pred,
                       const int4*   __restrict__ tgt,
                       uint32_t*     __restrict__ counts,
                       float4*       __restrict__ bce_out,
                       uint32_t*     __restrict__ bin_out) {
    __shared__ uint32_t hist[NCLS * BINS];
    for (int i = threadIdx.x; i < NCLS * BINS; i += blockDim.x) hist[i] = 0u;
    __syncthreads();

    const int stride = gridDim.x * blockDim.x;
    const int v0     = blockIdx.x * blockDim.x + threadIdx.x;
    int c0           = (4 * v0) % NCLS;            // one mod at loop entry ...
    const int dc     = (4 * stride) % NCLS;        // ... then carried incrementally
    for (int v = v0; v < NV; v += stride) {
        float4 p  = pred[v];
        int4   tg = tgt[v];
        __builtin_prefetch(&pred[v + stride], 0, 0);   // global_prefetch_b8
        __builtin_prefetch(&tgt[v + stride], 0, 0);

        float px[4] = {p.x, p.y, p.z, p.w};
        int   tx[4] = {tg.x, tg.y, tg.z, tg.w};
        float bce[4];
        uint32_t packed = 0u;
#pragma unroll
        for (int j = 0; j < 4; ++j) {
            float x  = px[j];
            float t  = (float)tx[j];
            float ax = fabsf(x);
            float e  = __expf(-ax);                 // exp(-|x|), reused by g and bce
            float inv1pe = 1.0f / (1.0f + e);
            float y  = (1.0f - 2.0f * t) * x;       // g = sigmoid(y)
            float g  = (y >= 0.0f) ? inv1pe : e * inv1pe;
            int   b  = (int)(g * (float)BINS);
            b = (b > BINS - 1) ? (BINS - 1) : b;
            bce[j] = fmaxf(x, 0.0f) - x * t + __logf(1.0f + e);
            atomicAdd(&hist[(c0 + j) * BINS + b], 1u);   // ds_add_u32
            packed |= ((uint32_t)b) << (8 * j);
        }
        if (CACHE) {
            bce_out[v] = make_float4(bce[0], bce[1], bce[2], bce[3]);
            bin_out[v] = packed;
        }
        c0 += dc; if (c0 >= NCLS) c0 -= NCLS;
    }

    __syncthreads();
    for (int i = threadIdx.x; i < NCLS * BINS; i += blockDim.x) {
        uint32_t h = hist[i];
        if (h) atomicAdd(&counts[i], h);                 // global_atomic_add_u32
    }
}

// ---------------- kernel C: EMA update + weight table ----------------
__global__ void k_weights(const uint32_t* __restrict__ counts,
                          const float*    __restrict__ acc_sum,
                          float*          __restrict__ wtab) {
    const int c   = blockIdx.x;      // one class per (single-wave) block
    const int tid = threadIdx.x;     // 32 threads
    __shared__ int s_n;
    if (tid == 0) s_n = 0;
    __syncthreads();

    float cnt = 0.0f, acc = 0.0f;
    bool  ne  = false;
    const bool valid = (tid < BINS);
    if (valid) {
        uint32_t cu = counts[c * BINS + tid];
        cnt = (float)cu;
        acc = acc_sum[c * BINS + tid];
        ne  = (cu > 0u);
        if (ne) atomicAdd(&s_n, 1);
    }
    __syncthreads();
    if (valid) {
        float na = ne ? (MMT * acc + (1.0f - MMT) * cnt) : acc;
        float w  = ne ? ((float)NBATCH / (na * (float)s_n)) : 0.0f;
        wtab[c * BINS + tid] = w;
    }
}

// ---------------- kernel D: weighted sum -> scalar ----------------
template <bool CACHE>
__global__ void k_loss(const float4*   __restrict__ pred,
                       const int4*     __restrict__ tgt,
                       const float4*   __restrict__ bce_in,
                       const uint32_t* __restrict__ bin_in,
                       const float*    __restrict__ wtab,
                       float*          __restrict__ out) {
    __shared__ float swt[NCLS * BINS];

    // Async-DMA the 4.8KB weight table into LDS: 300 x 16B chunks (ASYNCcnt path).
    {
        uint32_t base = (uint32_t)(uintptr_t)&swt[0];
        for (int i = threadIdx.x; i < (NCLS * BINS * 4) / 16; i += blockDim.x)
            async_lds_b128(base + (uint32_t)i * 16u, (const char*)wtab + (size_t)i * 16u);
        wait_async0();
    }
    __syncthreads();

    float sum = 0.0f;
    const int stride = gridDim.x * blockDim.x;
    const int v0     = blockIdx.x * blockDim.x + threadIdx.x;
    int c0           = (4 * v0) % NCLS;
    const int dc     = (4 * stride) % NCLS;
    for (int v = v0; v < NV; v += stride) {
        if (CACHE) {
            float4   b4 = bce_in[v];
            uint32_t pk = bin_in[v];
            __builtin_prefetch(&bce_in[v + stride], 0, 0);
            __builtin_prefetch(&bin_in[v + stride], 0, 0);
            sum += b4.x * swt[(c0 + 0) * BINS + (int)( pk        & 0xffu)];
            sum += b4.y * swt[(c0 + 1) * BINS + (int)((pk >> 8)  & 0xffu)];
            sum += b4.z * swt[(c0 + 2) * BINS + (int)((pk >> 16) & 0xffu)];
            sum += b4.w * swt[(c0 + 3) * BINS + (int)((pk >> 24) & 0xffu)];
        } else {
            float4 p  = pred[v];
            int4   tg = tgt[v];
            __builtin_prefetch(&pred[v + stride], 0, 0);
            __builtin_prefetch(&tgt[v + stride], 0, 0);
            float px[4] = {p.x, p.y, p.z, p.w};
            int   tx[4] = {tg.x, tg.y, tg.z, tg.w};
#pragma unroll
            for (int j = 0; j < 4; ++j) {
                float x  = px[j];
                float t  = (float)tx[j];
                float ax = fabsf(x);
                float e  = __expf(-ax);
                float inv1pe = 1.0f / (1.0f + e);
                float y  = (1.0f - 2.0f * t) * x;
                float g  = (y >= 0.0f) ? inv1pe : e * inv1pe;
                int   b  = (int)(g * (float)BINS);
                b = (b > BINS - 1) ? (BINS - 1) : b;
                float bce = fmaxf(x, 0.0f) - x * t + __logf(1.0f + e);
                sum += bce * swt[(c0 + j) * BINS + b];
            }
        }
        c0 += dc; if (c0 >= NCLS) c0 -= NCLS;
    }

    block_reduce_add(sum, out, 1.0f / (float)NTOT);
}

// ---------------- host ----------------
extern "C" void kernel_launch(void* const* d_in, const int* in_sizes, int n_in,
                              void* d_out, int out_size, void* d_ws, size_t ws_size,
                              hipStream_t stream) {
    (void)in_sizes; (void)n_in; (void)out_size;
    const float* pred = (const float*)d_in[0];
    const int*   tgt  = (const int*)d_in[1];
    const float* acc  = (const float*)d_in[2];
    float*       out  = (float*)d_out;

    char*     ws     = (char*)d_ws;
    uint32_t* counts = (uint32_t*)ws;
    float*    wtab   = (float*)(ws + 4800);

    const size_t need_cache = CACHE_OFF + (size_t)NTOT * 4 + (size_t)NTOT;
    const bool   cache      = (ws_size >= need_cache);
    float4*   bce_c = (float4*)(ws + CACHE_OFF);
    uint32_t* bin_c = (uint32_t*)(ws + CACHE_OFF + (size_t)NTOT * 4);

    k_zero<<<(NCLS * BINS + 255) / 256, 256, 0, stream>>>(counts, out);

    dim3 grid(2048), block(256);
    if (cache)
        k_hist<true ><<<grid, block, 0, stream>>>((const float4*)pred, (const int4*)tgt,
                                                  counts, bce_c, bin_c);
    else
        k_hist<false><<<grid, block, 0, stream>>>((const float4*)pred, (const int4*)tgt,
                                                  counts, nullptr, nullptr);

    k_weights<<<NCLS, 32, 0, stream>>>(counts, acc, wtab);

    if (cache)
        k_loss<true ><<<grid, block, 0, stream>>>((const float4*)pred, (const int4*)tgt,
                                                  bce_c, bin_c, wtab, out);
    else
        k_loss<false><<<grid, block, 0, stream>>>((const float4*)pred, (const int4*)tgt,
                                                  nullptr, nullptr, wtab, out);
}